// TemporalPrototypeManager_51427938402447
// MI455X (gfx1250) — compile-verified
//
#include <hip/hip_runtime.h>
#include <hip/hip_bf16.h>
#include <stdint.h>

// ---------------------------------------------------------------------------
// TemporalPrototypeManager on MI455X (gfx1250)
// segment-sum is HBM-bandwidth bound (~1.03 GB reads -> ~45us @ 23.3 TB/s).
// Strategy: per-block LDS-resident [1000 x 32] f32 accumulator (128 KB LDS,
// CDNA5-only capacity), features streamed via global_load_async_to_lds_b128
// with th:TH_LOAD_NT (read-once stream must not evict the L2-resident sums),
// accumulation via ds_add_f32, partials flushed with global_atomic_add_f32.
// ---------------------------------------------------------------------------

#define C_NUM      1000
#define D_DIM      128
#define N_SAMP     2000000
#define MOM        0.9f

#define GROUP_COLS 32                       // columns handled per block
#define NGROUPS    (D_DIM / GROUP_COLS)     // 4
#define RCHUNKS    250                      // sample chunks
#define CHUNK      (N_SAMP / RCHUNKS)       // 8000 (exact)
#define BLK        256
#define WAVES      (BLK / 32)               // 8
#define TILE_SAMP  32                       // samples per block-tile (8 waves * 4)
#define NTILES     (CHUNK / TILE_SAMP)      // 250 (exact)
#define DEPTH      4                        // async staging depth (per wave)

typedef __attribute__((address_space(3))) void lds_void_t;

// One b128 async per lane: 16B global -> 16B LDS, 512B per wave-instruction.
// Non-temporal: the 1 GB feature stream is read exactly once.
__device__ __forceinline__ void async_copy_b128_nt(const void* gptr, void* lptr) {
    asm volatile("global_load_async_to_lds_b128 %0, %1, off th:TH_LOAD_NT"
                 :
                 : "v"((lds_void_t*)lptr), "v"((uint64_t)(uintptr_t)gptr)
                 : "memory");
}

#define WAIT_ASYNC_LE(n) asm volatile("s_wait_asynccnt " #n ::: "memory")

// ---------------------------------------------------------------------------
__global__ void __launch_bounds__(BLK)
zero_ws_kernel(float* ws, int n) {
    int i = blockIdx.x * BLK + threadIdx.x;
    if (i < n) ws[i] = 0.0f;
}

// ---------------------------------------------------------------------------
__global__ void __launch_bounds__(BLK)
count_hist_kernel(const int* __restrict__ labels, float* __restrict__ counts) {
    __shared__ uint32_t h[C_NUM];
    for (int i = threadIdx.x; i < C_NUM; i += BLK) h[i] = 0u;
    __syncthreads();
    const int stride = gridDim.x * BLK;
    for (int i = blockIdx.x * BLK + threadIdx.x; i < N_SAMP; i += stride)
        atomicAdd(&h[__builtin_nontemporal_load(&labels[i])], 1u);
    __syncthreads();
    for (int i = threadIdx.x; i < C_NUM; i += BLK) {
        uint32_t v = h[i];
        if (v) atomicAdd(&counts[i], (float)v);
    }
}

// ---------------------------------------------------------------------------
__global__ void __launch_bounds__(BLK)
scatter_sum_kernel(const float* __restrict__ features,
                   const int*   __restrict__ labels,
                   float*       __restrict__ sums) {
    // 128000 B accumulator + 16384 B staging = 144384 B LDS
    __shared__ float s_accum[C_NUM * GROUP_COLS];
    __shared__ __align__(16) float s_stage[WAVES * DEPTH * 4 * GROUP_COLS];

    const int group = blockIdx.x & (NGROUPS - 1);
    const int rc    = blockIdx.x >> 2;
    const int c0    = group * GROUP_COLS;
    const int base  = rc * CHUNK;

    for (int i = threadIdx.x; i < C_NUM * GROUP_COLS; i += BLK) s_accum[i] = 0.0f;
    __syncthreads();

    const int lane = threadIdx.x & 31;
    const int wv   = threadIdx.x >> 5;
    const int ss   = lane >> 3;        // 0..3 : sub-sample within wave tile
    const int cq   = lane & 7;         // 0..7 : 4-col quad within 32-col slice

    // lane's fixed offsets
    const float* gbase =
        features + (size_t)(base + wv * 4 + ss) * D_DIM + c0 + cq * 4;
    float* stW = s_stage + wv * (DEPTH * 4 * GROUP_COLS);
    const int laneOff = ss * GROUP_COLS + cq * 4;

    // ---- prologue: issue DEPTH-1 tiles ----
#pragma unroll
    for (int t = 0; t < DEPTH - 1; ++t)
        async_copy_b128_nt(gbase + (size_t)t * TILE_SAMP * D_DIM,
                           stW + t * (4 * GROUP_COLS) + laneOff);

    int lab = __builtin_nontemporal_load(&labels[base + wv * 4 + ss]);

    const int steady = NTILES - (DEPTH - 1);
    int i = 0;
    for (; i < steady; ++i) {
        // keep DEPTH async loads in flight
        async_copy_b128_nt(gbase + (size_t)(i + DEPTH - 1) * TILE_SAMP * D_DIM,
                           stW + ((i + DEPTH - 1) & (DEPTH - 1)) * (4 * GROUP_COLS) + laneOff);
        // label for next tile (software-pipelined one tile ahead)
        int labN = __builtin_nontemporal_load(
            &labels[base + (i + 1) * TILE_SAMP + wv * 4 + ss]);

        WAIT_ASYNC_LE(3);   // DEPTH-1 : oldest tile landed in LDS

        const float4 v = *(const float4*)(stW + (i & (DEPTH - 1)) * (4 * GROUP_COLS) + laneOff);
        float* acc = &s_accum[lab * GROUP_COLS + cq * 4];
        atomicAdd(acc + 0, v.x);
        atomicAdd(acc + 1, v.y);
        atomicAdd(acc + 2, v.z);
        atomicAdd(acc + 3, v.w);
        lab = labN;
    }

    WAIT_ASYNC_LE(0);       // drain: last DEPTH-1 tiles are all in LDS
    for (; i < NTILES; ++i) {
        int labN = (i + 1 < NTILES)
                       ? __builtin_nontemporal_load(
                             &labels[base + (i + 1) * TILE_SAMP + wv * 4 + ss])
                       : 0;
        const float4 v = *(const float4*)(stW + (i & (DEPTH - 1)) * (4 * GROUP_COLS) + laneOff);
        float* acc = &s_accum[lab * GROUP_COLS + cq * 4];
        atomicAdd(acc + 0, v.x);
        atomicAdd(acc + 1, v.y);
        atomicAdd(acc + 2, v.z);
        atomicAdd(acc + 3, v.w);
        lab = labN;
    }

    __syncthreads();
    // flush block-private partials -> global sums (L2-resident, 512 KB)
    for (int idx = threadIdx.x; idx < C_NUM * GROUP_COLS; idx += BLK) {
        const int cls = idx >> 5;              // /GROUP_COLS
        const int col = idx & (GROUP_COLS - 1);
        atomicAdd(&sums[cls * D_DIM + c0 + col], s_accum[idx]);
    }
}

// ---------------------------------------------------------------------------
__global__ void __launch_bounds__(D_DIM)
finalize_kernel(const float* __restrict__ protos,
                const float* __restrict__ sums,
                const float* __restrict__ counts,
                float*       __restrict__ out) {
    const int c = blockIdx.x;
    const int d = threadIdx.x;   // 128 threads
    __shared__ float red[D_DIM];

    const float p = protos[c * D_DIM + d];
    red[d] = p;
    __syncthreads();
#pragma unroll
    for (int off = D_DIM / 2; off > 0; off >>= 1) {
        if (d < off) red[d] += red[d + off];
        __syncthreads();
    }
    const float rowsum = red[0];
    const float cnt    = counts[c];
    const float mean   = sums[c * D_DIM + d] / fmaxf(cnt, 1.0f);
    const float ema    = MOM * p + (1.0f - MOM) * mean;
    out[c * D_DIM + d] = (cnt > 0.0f) ? ((rowsum == 0.0f) ? mean : ema) : p;
}

// ---------------------------------------------------------------------------
extern "C" void kernel_launch(void* const* d_in, const int* in_sizes, int n_in,
                              void* d_out, int out_size, void* d_ws, size_t ws_size,
                              hipStream_t stream) {
    const float* features = (const float*)d_in[0];
    const int*   labels   = (const int*)d_in[1];
    const float* protos   = (const float*)d_in[2];
    float*       out      = (float*)d_out;

    float* sums   = (float*)d_ws;               // C*D floats
    float* counts = sums + C_NUM * D_DIM;       // C floats
    const int ws_elems = C_NUM * D_DIM + C_NUM; // 129000

    zero_ws_kernel<<<(ws_elems + BLK - 1) / BLK, BLK, 0, stream>>>(sums, ws_elems);
    count_hist_kernel<<<512, BLK, 0, stream>>>(labels, counts);
    scatter_sum_kernel<<<NGROUPS * RCHUNKS, BLK, 0, stream>>>(features, labels, sums);
    finalize_kernel<<<C_NUM, D_DIM, 0, stream>>>(protos, sums, counts, out);
}